// ADSREncoderV3_81028853006562
// MI455X (gfx1250) — compile-verified
//
#include <hip/hip_runtime.h>
#include <hip/hip_bf16.h>

#define BATCH  32
#define TF     4096
#define HOPSZ  512
#define PROTO  87
#define NMAXON 32
#define CEPS   1e-7f
#define BNEPS  1e-5f
#define NWG    (BATCH * (TF / 32))   // 4096 workgroups per pointwise layer

typedef float v2f __attribute__((ext_vector_type(2)));
typedef float v8f __attribute__((ext_vector_type(8)));

// ---------------------------------------------------------------------------
// Stage 1: per-frame RMS -> log.  One wave32 per 512-sample frame.
// ---------------------------------------------------------------------------
__global__ __launch_bounds__(256) void rms_kernel(const float* __restrict__ wav,
                                                  float* __restrict__ log_rms) {
  const int wave = threadIdx.x >> 5, lane = threadIdx.x & 31;
  const size_t frame = (size_t)blockIdx.x * 8 + wave;       // B*TF frames total
  const float4* p = (const float4*)wav + frame * (HOPSZ / 4);
  float s = 0.f;
#pragma unroll
  for (int j = 0; j < 4; ++j) {
    float4 v = p[lane + 32 * j];
    s += v.x * v.x + v.y * v.y + v.z * v.z + v.w * v.w;
  }
#pragma unroll
  for (int o = 16; o; o >>= 1) s += __shfl_xor(s, o, 32);
  if (lane == 0)
    log_rms[frame] = logf(sqrtf(s * (1.f / HOPSZ) + CEPS) + CEPS);
}

// ---------------------------------------------------------------------------
// Stage 2: stem conv (1->32, k=3, pad 1) + tanh-GELU (JAX approximate=True).
// One thread per time step, loops the 32 output channels (uniform weights ->
// scalar loads).
// ---------------------------------------------------------------------------
__global__ __launch_bounds__(256) void stem_kernel(const float* __restrict__ lr,
                                                   const float* __restrict__ w,
                                                   const float* __restrict__ bias,
                                                   float* __restrict__ x0) {
  const size_t i = (size_t)blockIdx.x * 256 + threadIdx.x;  // over B*TF
  const int b = (int)(i >> 12);
  const int t = (int)(i & (TF - 1));
  const float* l = lr + ((size_t)b * TF);
  const float xm = (t > 0) ? l[t - 1] : 0.f;
  const float xc = l[t];
  const float xp = (t < TF - 1) ? l[t + 1] : 0.f;
  float* o = x0 + (size_t)b * 32 * TF;
#pragma unroll 8
  for (int c = 0; c < 32; ++c) {
    float v = w[c * 3 + 0] * xm + w[c * 3 + 1] * xc + w[c * 3 + 2] * xp + bias[c];
    float u = 0.7978845608028654f * (v + 0.044715f * v * v * v);
    float e = __expf(2.f * u);
    float th = (e - 1.f) / (e + 1.f);
    o[(size_t)c * TF + t] = 0.5f * v * (1.f + th);
  }
}

// ---------------------------------------------------------------------------
// Stage 3: fused depthwise(k=3, dilation d) + pointwise(cin->64) conv using
// V_WMMA_F32_16X16X4_F32, plus deterministic per-workgroup BN partial sums.
//
// Workgroup (256 thr = 8 waves) handles one (batch, 32-frame tile).
//   A (16x4 f32): lanes 0-15 hold M=lane K={0,1}; lanes 16-31 M=lane-16 K={2,3}
//   B (4x16 f32): lanes 0-15 hold N=lane K={0,1}; lanes 16-31 N=lane-16 K={2,3}
//   C/D (16x16 f32): vgpr r, lanes 0-15 -> (M=r,N=lane); lanes 16-31 -> (M=r+8)
// ---------------------------------------------------------------------------
__global__ __launch_bounds__(256) void dwpw_kernel(
    const float* __restrict__ x,    // [B][cin][TF]
    const float* __restrict__ dw,   // [cin][3]
    const float* __restrict__ pw,   // [64][cin]
    float* __restrict__ y,          // [B][64][TF]
    float* __restrict__ psum,       // [NWG][64]
    float* __restrict__ psq,        // [NWG][64]
    int cin, int dil) {
  __shared__ float xs[64][48];   // input tile + 8-halo each side
  __shared__ float hs[64][40];   // depthwise result (stride 40 => +16 bank shift)
  __shared__ float pwT[64][72];  // pw transposed [ci][co] (stride 72 => +16 shift)
  __shared__ float dws[64 * 3];
  __shared__ float ps[2][64], ps2[2][64];

  const int wg = blockIdx.x;       // b*128 + ttile
  const int b = wg >> 7;
  const int t0 = (wg & 127) * 32;
  const int tid = threadIdx.x;

  for (int i = tid; i < cin * 3; i += 256) dws[i] = dw[i];
  for (int i = tid; i < 64 * cin; i += 256) {
    int co = i / cin, ci = i - co * cin;
    pwT[ci][co] = pw[i];
  }
  const float* xb = x + (size_t)b * cin * TF;
  for (int i = tid; i < cin * 48; i += 256) {
    int ci = i / 48, tt = i - ci * 48;
    int t = t0 + tt - 8;
    xs[ci][tt] = (t >= 0 && t < TF) ? xb[(size_t)ci * TF + t] : 0.f;
  }
  __syncthreads();

  for (int e = tid; e < cin * 32; e += 256) {
    int ci = e >> 5, tt = e & 31;
    float a0 = dws[ci * 3 + 0], a1 = dws[ci * 3 + 1], a2 = dws[ci * 3 + 2];
    hs[ci][tt] = a0 * xs[ci][tt + 8 - dil] + a1 * xs[ci][tt + 8] +
                 a2 * xs[ci][tt + 8 + dil];
  }
  __syncthreads();

  const int wave = tid >> 5, lane = tid & 31;
  const int cotile = wave >> 1, ttile = wave & 1;
  const int co_base = cotile * 16, tb = ttile * 16;
  const int lm = lane & 15;
  const bool hi = lane >= 16;

  v8f acc = {0.f, 0.f, 0.f, 0.f, 0.f, 0.f, 0.f, 0.f};
  const int kiters = cin >> 2;
  for (int k = 0; k < kiters; ++k) {
    const int k4 = k * 4 + (hi ? 2 : 0);
    v2f A, Bv;
    A.x = pwT[k4][co_base + lm];
    A.y = pwT[k4 + 1][co_base + lm];
    Bv.x = hs[k4][tb + lm];
    Bv.y = hs[k4 + 1][tb + lm];
    acc = __builtin_amdgcn_wmma_f32_16x16x4_f32(false, A, false, Bv,
                                                (short)0, acc, false, false);
  }

  float* yb = y + (size_t)b * 64 * TF;
#pragma unroll
  for (int r = 0; r < 8; ++r) {
    float v = acc[r];
    int co = co_base + r + (hi ? 8 : 0);
    yb[(size_t)co * TF + t0 + tb + lm] = v;
    float s = v, q = v * v;
#pragma unroll
    for (int o = 8; o; o >>= 1) {
      s += __shfl_xor(s, o, 32);
      q += __shfl_xor(q, o, 32);
    }
    if (lm == 0) { ps[ttile][co] = s; ps2[ttile][co] = q; }
  }
  __syncthreads();
  if (tid < 64) {
    psum[(size_t)wg * 64 + tid] = ps[0][tid] + ps[1][tid];
    psq[(size_t)wg * 64 + tid]  = ps2[0][tid] + ps2[1][tid];
  }
}

// ---------------------------------------------------------------------------
// Stage 4: BN statistics -> fused scale/shift (deterministic fixed-order tree)
// ---------------------------------------------------------------------------
__global__ __launch_bounds__(256) void stats_kernel(const float* __restrict__ psum,
                                                    const float* __restrict__ psq,
                                                    const float* __restrict__ g,
                                                    const float* __restrict__ bb,
                                                    float* __restrict__ scale,
                                                    float* __restrict__ shift) {
  const int co = blockIdx.x, tid = threadIdx.x;
  __shared__ float s1[256], s2[256];
  float a = 0.f, c = 0.f;
  for (int w = tid; w < NWG; w += 256) {
    a += psum[(size_t)w * 64 + co];
    c += psq[(size_t)w * 64 + co];
  }
  s1[tid] = a; s2[tid] = c;
  __syncthreads();
  for (int o = 128; o; o >>= 1) {
    if (tid < o) { s1[tid] += s1[tid + o]; s2[tid] += s2[tid + o]; }
    __syncthreads();
  }
  if (tid == 0) {
    const float N = (float)BATCH * (float)TF;
    float mean = s1[0] / N;
    float var = s2[0] / N - mean * mean;
    float sc = g[co] * rsqrtf(var + BNEPS);
    scale[co] = sc;
    shift[co] = bb[co] - mean * sc;
  }
}

// ---------------------------------------------------------------------------
// Stage 5: normalize + SiLU, in place.
// ---------------------------------------------------------------------------
__global__ __launch_bounds__(256) void norm_silu_kernel(float* __restrict__ y,
                                                        const float* __restrict__ scale,
                                                        const float* __restrict__ shift) {
  const size_t i = (size_t)blockIdx.x * 256 + threadIdx.x;  // B*64*TF
  const int c = (int)((i >> 12) & 63);
  float v = y[i] * scale[c] + shift[c];
  y[i] = v / (1.f + __expf(-v));
}

// ---------------------------------------------------------------------------
// Stage 6: first NMAX onset positions per batch (ballot scan), lengths, weights
// ---------------------------------------------------------------------------
__global__ __launch_bounds__(32) void onset_kernel(const int* __restrict__ flags,
                                                   int* __restrict__ onpos,
                                                   float* __restrict__ wgt,
                                                   float* __restrict__ lenf) {
  const int b = blockIdx.x, lane = threadIdx.x;
  const int* f = flags + (size_t)b * TF;
  __shared__ int pos[NMAXON];
  if (lane < NMAXON) pos[lane] = TF;
  __syncthreads();
  int count = 0;
  for (int ch = 0; ch < TF / 32 && count < NMAXON; ++ch) {
    int t = ch * 32 + lane;
    bool on = (f[t] == 1);
    unsigned mask = (unsigned)__ballot(on);
    if (on) {
      int ix = count + __popc(mask & ((1u << lane) - 1u));
      if (ix < NMAXON) pos[ix] = t;
    }
    count += __popc(mask);
  }
  __syncthreads();
  int p = pos[lane];
  int nx = (lane < NMAXON - 1) ? pos[lane + 1] : TF;
  int len = nx - p;
  len = (len < 0) ? 0 : ((len > PROTO) ? PROTO : len);
  float fl = (float)len, tot = fl;
#pragma unroll
  for (int o = 16; o; o >>= 1) tot += __shfl_xor(tot, o, 32);
  onpos[b * NMAXON + lane] = p;
  wgt[b * NMAXON + lane] = fl / (tot + CEPS);
  lenf[b * NMAXON + lane] = fl;
}

// ---------------------------------------------------------------------------
// Stage 7: proto_E[b][c][p] = sum_n w_n * feat[b][c][pos_n + p] (masked)
// ---------------------------------------------------------------------------
__global__ __launch_bounds__(256) void proto_kernel(const float* __restrict__ feat,
                                                    const int* __restrict__ onpos,
                                                    const float* __restrict__ wgt,
                                                    const float* __restrict__ lenf,
                                                    float* __restrict__ proto) {
  const int b = blockIdx.x;
  __shared__ int sp[NMAXON], sl[NMAXON];
  __shared__ float sw[NMAXON];
  if (threadIdx.x < NMAXON) {
    sp[threadIdx.x] = onpos[b * NMAXON + threadIdx.x];
    sw[threadIdx.x] = wgt[b * NMAXON + threadIdx.x];
    sl[threadIdx.x] = (int)lenf[b * NMAXON + threadIdx.x];
  }
  __syncthreads();
  const float* fb = feat + (size_t)b * 64 * TF;
  for (int e = threadIdx.x; e < 64 * PROTO; e += 256) {
    int c = e / PROTO, p = e - c * PROTO;
    float acc = 0.f;
    for (int n = 0; n < NMAXON; ++n) {
      int ptn = sp[n];
      if (ptn >= TF) break;           // sorted: rest are padding
      if (p < sl[n]) {
        int idx = ptn + p;
        if (idx > TF - 1) idx = TF - 1;
        acc += sw[n] * fb[(size_t)c * TF + idx];
      }
    }
    proto[((size_t)b * 64 + c) * PROTO + p] = acc;
  }
}

// ---------------------------------------------------------------------------
// Stage 8: cummax of onset position -> offset since last onset (segmented scan)
// ---------------------------------------------------------------------------
__global__ __launch_bounds__(256) void lastoff_kernel(const int* __restrict__ flags,
                                                      int* __restrict__ offarr) {
  const int b = blockIdx.x;
  const int* f = flags + (size_t)b * TF;
  __shared__ int segmax[256], pre[257];
  const int tid = threadIdx.x, base = tid * 16;
  int m = -1;
  for (int j = 0; j < 16; ++j)
    if (f[base + j] == 1) m = base + j;
  segmax[tid] = m;
  __syncthreads();
  if (tid == 0) {
    int c = -1;
    for (int j = 0; j < 256; ++j) { pre[j] = c; if (segmax[j] > c) c = segmax[j]; }
    pre[256] = c;
  }
  __syncthreads();
  int carry = pre[tid];
  for (int j = 0; j < 16; ++j) {
    int t = base + j;
    if (f[t] == 1) carry = t;
    int off = t - carry;
    offarr[(size_t)b * TF + t] = (carry >= 0 && off < PROTO) ? off : -1;
  }
}

// ---------------------------------------------------------------------------
// Stage 9: adsr_stream[b][c][t] = proto_E[b][c][off] where valid else 0
// ---------------------------------------------------------------------------
__global__ __launch_bounds__(256) void adsr_kernel(const float* __restrict__ proto,
                                                   const int* __restrict__ offarr,
                                                   float* __restrict__ outp) {
  const int bc = blockIdx.x;               // b*64 + c
  const int b = bc >> 6;
  __shared__ float pr[PROTO];
  if (threadIdx.x < PROTO) pr[threadIdx.x] = proto[(size_t)bc * PROTO + threadIdx.x];
  __syncthreads();
  const int* ob = offarr + (size_t)b * TF;
  float* o = outp + (size_t)bc * TF;
  for (int t = threadIdx.x; t < TF; t += 256) {
    int off = ob[t];
    o[t] = (off >= 0) ? pr[off] : 0.f;
  }
}

// ---------------------------------------------------------------------------
extern "C" void kernel_launch(void* const* d_in, const int* in_sizes, int n_in,
                              void* d_out, int out_size, void* d_ws, size_t ws_size,
                              hipStream_t stream) {
  (void)n_in; (void)out_size; (void)ws_size;

  // Input index mapping: detect dict-insertion order vs. alphabetical
  // tree-flatten order using in_sizes[0] (wav has 32*2097152 elements).
  int iw, ifl, iwst, ibst, idw[4], ipw[4], ig[4], ib[4];
  if (in_sizes[0] == BATCH * TF * HOPSZ) {   // insertion order
    iw = 0; ifl = 1; iwst = 2; ibst = 3;
    for (int i = 0; i < 4; ++i) {
      idw[i] = 4 + 4 * i; ipw[i] = 5 + 4 * i; ig[i] = 6 + 4 * i; ib[i] = 7 + 4 * i;
    }
  } else {                                   // alphabetical: b0..b3,b_stem,dw0..3,g0..3,pw0..3,w_stem
    ifl = 0;
    for (int i = 0; i < 4; ++i) { ib[i] = 1 + i; idw[i] = 6 + i; ig[i] = 10 + i; ipw[i] = 14 + i; }
    ibst = 5; iwst = 18; iw = 19;
  }

  const float* wav   = (const float*)d_in[iw];
  const int*   flags = (const int*)d_in[ifl];
  const float* wstem = (const float*)d_in[iwst];
  const float* bstem = (const float*)d_in[ibst];

  // Workspace layout (float-element offsets)
  float* ws = (float*)d_ws;
  const size_t O_LOGRMS = 0;                                  // 131072
  const size_t O_X0     = O_LOGRMS + (size_t)BATCH * TF;      // 4,194,304
  const size_t O_BUFA   = O_X0 + (size_t)BATCH * 32 * TF;     // 8,388,608
  const size_t O_BUFB   = O_BUFA + (size_t)BATCH * 64 * TF;
  const size_t O_PSUM   = O_BUFB + (size_t)BATCH * 64 * TF;   // NWG*64
  const size_t O_PSQ    = O_PSUM + (size_t)NWG * 64;
  const size_t O_SCALE  = O_PSQ + (size_t)NWG * 64;           // 64
  const size_t O_SHIFT  = O_SCALE + 64;
  const size_t O_ONPOS  = O_SHIFT + 64;                       // 1024 ints
  const size_t O_WGT    = O_ONPOS + BATCH * NMAXON;
  const size_t O_LEN    = O_WGT + BATCH * NMAXON;
  const size_t O_OFF    = O_LEN + BATCH * NMAXON;             // B*TF ints

  float* log_rms = ws + O_LOGRMS;
  float* x0      = ws + O_X0;
  float* bufA    = ws + O_BUFA;
  float* bufB    = ws + O_BUFB;
  float* psum    = ws + O_PSUM;
  float* psq     = ws + O_PSQ;
  float* scale   = ws + O_SCALE;
  float* shift   = ws + O_SHIFT;
  int*   onpos   = (int*)(ws + O_ONPOS);
  float* wgt     = ws + O_WGT;
  float* lenf    = ws + O_LEN;
  int*   offarr  = (int*)(ws + O_OFF);

  float* proto = (float*)d_out;                               // [B][64][PROTO]
  float* adsr  = proto + (size_t)BATCH * 64 * PROTO;          // [B][64][TF]

  // 1) RMS + log
  rms_kernel<<<(BATCH * TF) / 8, 256, 0, stream>>>(wav, log_rms);
  // 2) stem + GELU
  stem_kernel<<<(BATCH * TF) / 256, 256, 0, stream>>>(log_rms, wstem, bstem, x0);

  // 3) four dilated blocks: ping-pong x0 -> A -> B -> A -> B
  const int dils[4] = {1, 2, 4, 8};
  const float* bin[4]  = {x0, bufA, bufB, bufA};
  float*       bout[4] = {bufA, bufB, bufA, bufB};
  const int    cins[4] = {32, 64, 64, 64};
  for (int i = 0; i < 4; ++i) {
    const float* dwp = (const float*)d_in[idw[i]];
    const float* pwp = (const float*)d_in[ipw[i]];
    const float* gp  = (const float*)d_in[ig[i]];
    const float* bp  = (const float*)d_in[ib[i]];
    dwpw_kernel<<<NWG, 256, 0, stream>>>(bin[i], dwp, pwp, bout[i],
                                         psum, psq, cins[i], dils[i]);
    stats_kernel<<<64, 256, 0, stream>>>(psum, psq, gp, bp, scale, shift);
    norm_silu_kernel<<<(BATCH * 64 * TF) / 256, 256, 0, stream>>>(bout[i], scale, shift);
  }
  const float* feat = bufB;

  // 4) onsets -> proto_E -> adsr_stream
  onset_kernel<<<BATCH, 32, 0, stream>>>(flags, onpos, wgt, lenf);
  proto_kernel<<<BATCH, 256, 0, stream>>>(feat, onpos, wgt, lenf, proto);
  lastoff_kernel<<<BATCH, 256, 0, stream>>>(flags, offarr);
  adsr_kernel<<<BATCH * 64, 256, 0, stream>>>(proto, offarr, adsr);
}